// GCRNN_21629455303102
// MI455X (gfx1250) — compile-verified
//
#include <hip/hip_runtime.h>
#include <hip/hip_bf16.h>

#define AS1 __attribute__((address_space(1)))
#define AS3 __attribute__((address_space(3)))

#if defined(__gfx1250__) && __has_builtin(__builtin_amdgcn_global_load_async_to_lds_b128) && __has_builtin(__builtin_amdgcn_s_wait_asynccnt)
#define LGCN_ASYNC 1
#else
#define LGCN_ASYNC 0
#endif

#define EMB 128
#define NCATS 20
#define WAVES_PER_BLK 8

// 16-byte payload type for the b128 async-copy builtin
typedef int lgcn_v4i __attribute__((vector_size(16)));

__device__ __forceinline__ int lgcn_uniform(int x) {
#if defined(__gfx1250__)
    return __builtin_amdgcn_readfirstlane(x);   // wave-uniform -> SGPR -> s_load path
#else
    return x;
#endif
}

// ---------------- small utility kernels ----------------

__global__ void lgcn_zero_i32(int* __restrict__ p, int n) {
    int i = blockIdx.x * blockDim.x + threadIdx.x;
    if (i < n) p[i] = 0;
}

// n4 = count of float4 elements
__global__ void lgcn_zero_f32v4(float4* __restrict__ p, long long n4) {
    long long i = (long long)blockIdx.x * blockDim.x + threadIdx.x;
    if (i < n4) p[i] = make_float4(0.f, 0.f, 0.f, 0.f);
}

// in-degree count of dst (recv_mask == cnt>0, deg == max(cnt,1))
__global__ void lgcn_count(const int* __restrict__ dst, int* __restrict__ cnt, int E) {
    int i = blockIdx.x * blockDim.x + threadIdx.x;
    if (i < E) atomicAdd(&cnt[dst[i]], 1);
}

// per-edge coefficient w[e] = rsqrt(deg[src]*deg[dst])
__global__ void lgcn_edge_w(const int* __restrict__ src, const int* __restrict__ dst,
                            const int* __restrict__ cnt, float* __restrict__ w, int E) {
    int i = blockIdx.x * blockDim.x + threadIdx.x;
    if (i < E) {
        float ds = fmaxf((float)cnt[src[i]], 1.0f);
        float dd = fmaxf((float)cnt[dst[i]], 1.0f);
        w[i] = rsqrtf(ds * dd);
    }
}

// h_cur = h ; acc = alpha[0] * h   (float4 streaming, B128 ops)
__global__ void lgcn_init(const float4* __restrict__ h, const float* __restrict__ alpha,
                          float4* __restrict__ h_cur, float4* __restrict__ acc, long long n4) {
    long long i = (long long)blockIdx.x * blockDim.x + threadIdx.x;
    if (i < n4) {
        float a0 = alpha[0];
        float4 v = h[i];
        h_cur[i] = v;
        acc[i] = make_float4(a0 * v.x, a0 * v.y, a0 * v.z, a0 * v.w);
    }
}

// h_new = recv ? agg : h_cur ; h_cur = h_new ; acc += alpha[k+1] * h_new
// float4 covers 4 columns of one node (EMB=128, 32 float4 per row) -> mask uniform per vector
__global__ void lgcn_update(const int* __restrict__ cnt, const float* __restrict__ alpha,
                            const float4* __restrict__ agg, float4* __restrict__ h_cur,
                            float4* __restrict__ acc, long long n4, int k) {
    long long i = (long long)blockIdx.x * blockDim.x + threadIdx.x;
    if (i < n4) {
        int node = (int)(i >> 5);  // (i*4) >> 7
        float ak = alpha[k + 1];
        float4 v = (cnt[node] > 0) ? agg[i] : h_cur[i];
        h_cur[i] = v;
        float4 a = acc[i];
        acc[i] = make_float4(a.x + ak * v.x, a.y + ak * v.y,
                             a.z + ak * v.z, a.w + ak * v.w);
    }
}

// ---------------- heavy kernel: gather * scale -> atomic scatter ----------------
// One wave per edge per iteration; lane l owns columns [4l, 4l+4).
// Edge metadata is wave-uniform -> readfirstlane -> scalar (SMEM) loads.
// h[src] rows gathered via double-buffered async-copy-to-LDS (ASYNCcnt pipeline).
__global__ __launch_bounds__(256) void lgcn_msg(
    const float* __restrict__ h, const float* __restrict__ rel,
    const float* __restrict__ w, const int* __restrict__ src,
    const int* __restrict__ dst, const int* __restrict__ cat,
    float* __restrict__ agg, int E)
{
    __shared__ float s_rel[NCATS * EMB];  // 10 KB: category embeddings, hot
    for (int i = threadIdx.x; i < NCATS * EMB; i += blockDim.x) s_rel[i] = rel[i];
    __syncthreads();

    const int lane  = threadIdx.x & 31;
    const int wslot = threadIdx.x >> 5;
    const int gw = blockIdx.x * WAVES_PER_BLK + wslot;
    const int nw = gridDim.x * WAVES_PER_BLK;

#if LGCN_ASYNC
    __shared__ float4 s_h[2][WAVES_PER_BLK][32];
    int e = gw;
    if (e < E) {
        const int su = lgcn_uniform(src[lgcn_uniform(e)]);
        const float* gp = h + (size_t)su * EMB + lane * 4;
        __builtin_amdgcn_global_load_async_to_lds_b128(
            (AS1 lgcn_v4i*)gp, (AS3 lgcn_v4i*)&s_h[0][wslot][lane], 0, 0);
    }
    int par = 0;
    for (; e < E; e += nw) {
        const int en = e + nw;
        if (en < E) {
            const int su = lgcn_uniform(src[lgcn_uniform(en)]);
            const float* gp = h + (size_t)su * EMB + lane * 4;
            __builtin_amdgcn_global_load_async_to_lds_b128(
                (AS1 lgcn_v4i*)gp, (AS3 lgcn_v4i*)&s_h[par ^ 1][wslot][lane], 0, 0);
            __builtin_amdgcn_s_wait_asynccnt(1);  // in-order: current buffer ready
        } else {
            __builtin_amdgcn_s_wait_asynccnt(0);
        }
        const int   eu = lgcn_uniform(e);
        const int   d  = lgcn_uniform(dst[eu]);
        const int   c  = lgcn_uniform(cat[eu]);
        const float we = __uint_as_float(lgcn_uniform(__float_as_uint(w[eu])));
        const float4 hv = s_h[par][wslot][lane];
        const float4 r  = *(const float4*)&s_rel[c * EMB + lane * 4];
        float* op = agg + (size_t)d * EMB + lane * 4;
        atomicAdd(op + 0, hv.x * r.x * we);
        atomicAdd(op + 1, hv.y * r.y * we);
        atomicAdd(op + 2, hv.z * r.z * we);
        atomicAdd(op + 3, hv.w * r.w * we);
        par ^= 1;
    }
#else
    for (int e = gw; e < E; e += nw) {
        const int en = e + nw;
        if (en < E) {
            __builtin_prefetch(h + (size_t)src[lgcn_uniform(en)] * EMB + lane * 4, 0, 1);
        }
        const int   eu = lgcn_uniform(e);
        const int   s  = lgcn_uniform(src[eu]);
        const int   d  = lgcn_uniform(dst[eu]);
        const int   c  = lgcn_uniform(cat[eu]);
        const float we = __uint_as_float(lgcn_uniform(__float_as_uint(w[eu])));
        const float4 hv = *(const float4*)(h + (size_t)s * EMB + lane * 4);
        const float4 r  = *(const float4*)&s_rel[c * EMB + lane * 4];
        float* op = agg + (size_t)d * EMB + lane * 4;
        atomicAdd(op + 0, hv.x * r.x * we);
        atomicAdd(op + 1, hv.y * r.y * we);
        atomicAdd(op + 2, hv.z * r.z * we);
        atomicAdd(op + 3, hv.w * r.w * we);
    }
#endif
}

// ---------------- launcher ----------------

extern "C" void kernel_launch(void* const* d_in, const int* in_sizes, int n_in,
                              void* d_out, int out_size, void* d_ws, size_t ws_size,
                              hipStream_t stream) {
    const float* h     = (const float*)d_in[0];
    const float* rel   = (const float*)d_in[1];
    const float* alpha = (const float*)d_in[2];
    const int*   src   = (const int*)d_in[3];
    const int*   dst   = (const int*)d_in[4];
    const int*   cat   = (const int*)d_in[5];
    float* out = (float*)d_out;

    const int N = in_sizes[0] / EMB;        // 200000
    const int E = in_sizes[3];              // 1600000
    const int K = in_sizes[2] - 1;          // 3 hops
    const long long ND  = (long long)N * EMB;
    const long long ND4 = ND / 4;

    // workspace layout (512B aligned slices)
    auto align512 = [](size_t x) { return (x + 511) & ~(size_t)511; };
    char* ws = (char*)d_ws;
    size_t off = 0;
    int*   cnt   = (int*)(ws + off);   off += align512((size_t)N * 4);
    float* w     = (float*)(ws + off); off += align512((size_t)E * 4);
    float* h_cur = (float*)(ws + off); off += align512((size_t)ND * 4);
    float* agg   = (float*)(ws + off); off += align512((size_t)ND * 4);
    (void)ws_size; (void)n_in; (void)out_size;

    const int T = 256;
    const int bN   = (N + T - 1) / T;
    const int bE   = (E + T - 1) / T;
    const int bND4 = (int)((ND4 + T - 1) / T);
    const int bMSG = 8192;  // grid-stride: 65536 waves over 1.6M edges

    lgcn_zero_i32<<<bN, T, 0, stream>>>(cnt, N);
    lgcn_count<<<bE, T, 0, stream>>>(dst, cnt, E);
    lgcn_edge_w<<<bE, T, 0, stream>>>(src, dst, cnt, w, E);
    lgcn_init<<<bND4, T, 0, stream>>>((const float4*)h, alpha,
                                      (float4*)h_cur, (float4*)out, ND4);

    for (int k = 0; k < K; ++k) {
        lgcn_zero_f32v4<<<bND4, T, 0, stream>>>((float4*)agg, ND4);
        lgcn_msg<<<bMSG, T, 0, stream>>>(h_cur, rel, w, src, dst, cat, agg, E);
        lgcn_update<<<bND4, T, 0, stream>>>(cnt, alpha, (const float4*)agg,
                                            (float4*)h_cur, (float4*)out, ND4, k);
    }
}